// Model_2929167696455
// MI455X (gfx1250) — compile-verified
//
#include <hip/hip_runtime.h>
#include <hip/hip_bf16.h>

typedef float v2f __attribute__((ext_vector_type(2)));
typedef float v8f __attribute__((ext_vector_type(8)));

#define ND_ 20000
#define NS_ 8000
#define NT_ 4000
#define DD_ 128
#define NSYM_ 4
#define BB_ 64
#define E1_ 400000
#define E2_ 200000

static inline int cdiv(long a, long b) { return (int)((a + b - 1) / b); }

// ---------------- utility kernels (float4, counts are multiples of 4) ----------------
__global__ void zero4_kernel(float4* __restrict__ p, long n4) {
    long i = (long)blockIdx.x * 256 + threadIdx.x;
    if (i < n4) p[i] = make_float4(0.f, 0.f, 0.f, 0.f);
}

__global__ void add4_kernel(float4* __restrict__ acc, const float4* __restrict__ x, long n4) {
    long i = (long)blockIdx.x * 256 + threadIdx.x;
    if (i < n4) {
        float4 a = acc[i], b = x[i];
        acc[i] = make_float4(a.x + b.x, a.y + b.y, a.z + b.z, a.w + b.w);
    }
}

__global__ void scale4_kernel(float4* __restrict__ p, float f, long n4) {
    long i = (long)blockIdx.x * 256 + threadIdx.x;
    if (i < n4) {
        float4 a = p[i];
        p[i] = make_float4(a.x * f, a.y * f, a.z * f, a.w * f);
    }
}

__global__ void zero_kernel(float* __restrict__ p, long n) {
    long i = (long)blockIdx.x * 256 + threadIdx.x;
    if (i < n) p[i] = 0.f;
}

__global__ void deg_kernel(const int* __restrict__ src, const int* __restrict__ dst,
                           int E, int nsrc, float* __restrict__ deg) {
    int e = blockIdx.x * 256 + threadIdx.x;
    if (e < E) {
        atomicAdd(&deg[src[e]], 1.f);
        atomicAdd(&deg[nsrc + dst[e]], 1.f);
    }
}

__global__ void dinv_kernel(float* __restrict__ deg, int n) {
    int i = blockIdx.x * 256 + threadIdx.x;
    if (i < n) {
        float d = deg[i];
        deg[i] = (d > 0.f) ? rsqrtf(fmaxf(d, 1.f)) : 0.f;
    }
}

// one wave (32 lanes) per directed edge; lane handles 4 contiguous floats
__global__ void prop_kernel(const int* __restrict__ src, const int* __restrict__ dst,
                            int E, int nsrc, const float* __restrict__ dinv,
                            const float* __restrict__ ek, float* __restrict__ out) {
    long gid = (long)blockIdx.x * blockDim.x + threadIdx.x;
    long widx = gid >> 5;
    int lane = (int)(gid & 31);
    if (widx >= 2L * E) return;
    int r, c;
    if (widx < E) { r = src[widx]; c = nsrc + dst[widx]; }
    else { long e = widx - E; r = nsrc + dst[e]; c = src[e]; }
    float norm = dinv[r] * dinv[c];
    const float4 v = *((const float4*)(ek + (size_t)c * DD_) + lane);
    float* op = out + (size_t)r * DD_ + lane * 4;
    atomicAdd(op + 0, norm * v.x);
    atomicAdd(op + 1, norm * v.y);
    atomicAdd(op + 2, norm * v.z);
    atomicAdd(op + 3, norm * v.w);
}

// ---------------- fused FC GEMM: Y = X@W + bias + emb  (N = 128, M % 16 == 0) ----------------
// K compile-time: guard-free main loop; single guarded tail WMMA only when K % 4 != 0.
template <int KC>
__global__ void fc_gemm_kernel(const float* __restrict__ X,
                               const float* __restrict__ W,
                               const float* __restrict__ bias,
                               const float* __restrict__ emb,
                               float* __restrict__ Y) {
    int m0 = blockIdx.x * 16;
    int wave = threadIdx.x >> 5, lane = threadIdx.x & 31;
    int half = lane >> 4, lr = lane & 15;
    int n = wave * 16 + lr;
    const float* ap = X + (size_t)(m0 + lr) * KC;
    const float* bp = W + n;
    v8f c = {};
    constexpr int K4 = KC & ~3;
#pragma unroll 8
    for (int k0 = 0; k0 < K4; k0 += 4) {
        int ka = k0 + 2 * half;              // A/B K index: reg + 2*half
        v2f a, b;
        a.x = ap[ka];
        a.y = ap[ka + 1];
        b.x = bp[(size_t)ka * DD_];
        b.y = bp[(size_t)(ka + 1) * DD_];
        c = __builtin_amdgcn_wmma_f32_16x16x4_f32(false, a, false, b, (short)0, c, false, false);
    }
    if constexpr ((KC & 3) != 0) {
        int ka = K4 + 2 * half;
        v2f a, b;
        a.x = (ka     < KC) ? ap[ka]     : 0.f;
        a.y = (ka + 1 < KC) ? ap[ka + 1] : 0.f;
        b.x = (ka     < KC) ? bp[(size_t)ka * DD_]       : 0.f;
        b.y = (ka + 1 < KC) ? bp[(size_t)(ka + 1) * DD_] : 0.f;
        c = __builtin_amdgcn_wmma_f32_16x16x4_f32(false, a, false, b, (short)0, c, false, false);
    }
    float bn = bias[n];
#pragma unroll
    for (int g = 0; g < 8; ++g) {
        size_t rr = (size_t)(m0 + g + 8 * half);   // C/D: M = reg + 8*half
        Y[rr * DD_ + n] = c[g] + bn + emb[rr * DD_ + n];
    }
}

// ---------------- gate: des[s] = z_dis * sigmoid(z_dis@Wg[s]+bg[s]); scores fused ----------------
__global__ void gate_kernel(const float* __restrict__ z_dis, const float* __restrict__ Wg,
                            const float* __restrict__ bg, const float* __restrict__ att_vec,
                            float* __restrict__ des, float* __restrict__ scores) {
    int s = blockIdx.y;
    int m0 = blockIdx.x * 16;                // ND_ multiple of 16
    const float* W = Wg + (size_t)s * DD_ * DD_;
    __shared__ float sscore[16];
    if (threadIdx.x < 16) sscore[threadIdx.x] = 0.f;
    __syncthreads();
    int wave = threadIdx.x >> 5, lane = threadIdx.x & 31;
    int half = lane >> 4, lr = lane & 15;
    int n = wave * 16 + lr;
    const float* ap = z_dis + (size_t)(m0 + lr) * DD_;
    const float* bp = W + n;
    v8f c = {};
#pragma unroll 8
    for (int k0 = 0; k0 < DD_; k0 += 4) {
        int ka = k0 + 2 * half;
        v2f a, b;
        a.x = ap[ka];
        a.y = ap[ka + 1];
        b.x = bp[(size_t)ka * DD_];
        b.y = bp[(size_t)(ka + 1) * DD_];
        c = __builtin_amdgcn_wmma_f32_16x16x4_f32(false, a, false, b, (short)0, c, false, false);
    }
    float av = att_vec[n];
    float bb = bg[s * DD_ + n];
#pragma unroll
    for (int g = 0; g < 8; ++g) {
        size_t rr = (size_t)(m0 + g + 8 * half);
        // fast sigmoid: single v_rcp_f32 instead of full IEEE divide sequence
        float gate = __builtin_amdgcn_rcpf(1.f + __expf(-(c[g] + bb)));
        float dv = z_dis[rr * DD_ + n] * gate;
        des[((size_t)s * ND_ + rr) * DD_ + n] = dv;
        atomicAdd(&sscore[g + 8 * half], dv * av);   // ds_add_f32
    }
    __syncthreads();
    if (threadIdx.x < 16) scores[(size_t)s * ND_ + m0 + threadIdx.x] = sscore[threadIdx.x];
}

// ---------------- masked softmax stats per (b,s); stores max and 1/sum ----------------
__global__ void stats_kernel(const float* __restrict__ scores, const int* __restrict__ dm,
                             float* __restrict__ smax, float* __restrict__ sinv) {
    int b = blockIdx.x, s = blockIdx.y;
    const float* sc = scores + (size_t)s * ND_;
    const int* m = dm + (size_t)b * ND_;
    __shared__ float red[256];
    float mx = -1e9f;
    for (int n = threadIdx.x; n < ND_; n += 256)
        mx = fmaxf(mx, m[n] ? sc[n] : -1e9f);
    red[threadIdx.x] = mx; __syncthreads();
    for (int o = 128; o > 0; o >>= 1) {
        if (threadIdx.x < o) red[threadIdx.x] = fmaxf(red[threadIdx.x], red[threadIdx.x + o]);
        __syncthreads();
    }
    mx = red[0]; __syncthreads();
    float sum = 0.f;
    for (int n = threadIdx.x; n < ND_; n += 256)
        sum += __expf((m[n] ? sc[n] : -1e9f) - mx);
    red[threadIdx.x] = sum; __syncthreads();
    for (int o = 128; o > 0; o >>= 1) {
        if (threadIdx.x < o) red[threadIdx.x] += red[threadIdx.x + o];
        __syncthreads();
    }
    if (threadIdx.x == 0) {
        smax[s * BB_ + b] = mx;
        sinv[s * BB_ + b] = 1.f / red[0];   // 256 full-precision divides total
    }
}

__global__ void wmat_kernel(const float* __restrict__ scores, const int* __restrict__ dm,
                            const float* __restrict__ smax, const float* __restrict__ sinv,
                            float* __restrict__ Wm) {
    long idx = (long)blockIdx.x * 256 + threadIdx.x;
    if (idx >= (long)NSYM_ * BB_ * ND_) return;
    int n = (int)(idx % ND_);
    int b = (int)((idx / ND_) % BB_);
    int s = (int)(idx / ((long)ND_ * BB_));
    float v = dm[(size_t)b * ND_ + n] ? scores[(size_t)s * ND_ + n] : -1e9f;
    Wm[idx] = __expf(v - smax[s * BB_ + b]) * sinv[s * BB_ + b];
}

// ---------------- b_sym[b][s][:] = Wm[s] (64 x 20000) @ des[s] (20000 x 128) ----------------
__global__ void bsym_gemm_kernel(const float* __restrict__ Wm, const float* __restrict__ des,
                                 float* __restrict__ bsym) {
    int s = blockIdx.y;
    int m0 = blockIdx.x * 16;                 // M = 64
    const float* A = Wm + (size_t)s * BB_ * ND_;
    const float* B = des + (size_t)s * ND_ * DD_;
    int wave = threadIdx.x >> 5, lane = threadIdx.x & 31;
    int half = lane >> 4, lr = lane & 15;
    int n = wave * 16 + lr;
    const float* ap = A + (size_t)(m0 + lr) * ND_;
    const float* bp = B + n;
    v8f c = {};
    for (int kk = 0; kk < ND_; kk += 80) {    // 20000 = 250 * 80
        __builtin_prefetch(ap + kk + 80, 0, 0);   // stream A ahead (global_prefetch_b8)
#pragma unroll
        for (int k0 = kk; k0 < kk + 80; k0 += 4) {
            int ka = k0 + 2 * half;
            v2f a, b;
            a.x = ap[ka];
            a.y = ap[ka + 1];
            b.x = bp[(size_t)ka * DD_];
            b.y = bp[(size_t)(ka + 1) * DD_];
            c = __builtin_amdgcn_wmma_f32_16x16x4_f32(false, a, false, b, (short)0, c, false, false);
        }
    }
#pragma unroll
    for (int g = 0; g < 8; ++g) {
        int rr = m0 + g + 8 * half;           // batch index
        bsym[((size_t)rr * NSYM_ + s) * DD_ + n] = c[g];
    }
}

// ---------------- tail: v,o,Wl,sym_agg softmax -> syn_out [64,128] ----------------
__global__ void tail_kernel(const float* __restrict__ bsym,
                            const float* __restrict__ Wv, const float* __restrict__ bv,
                            const float* __restrict__ Wo, const float* __restrict__ bo,
                            const float* __restrict__ Wl, const float* __restrict__ bl,
                            const float* __restrict__ symW, const float* __restrict__ symb,
                            const float* __restrict__ symu, float* __restrict__ syn_out) {
    int b = blockIdx.x;
    int d = threadIdx.x;                       // 128 threads
    __shared__ float bufA[128], bufB[128], ss[NSYM_][128], red[128], e2[NSYM_];
    for (int s = 0; s < NSYM_; ++s) {
        bufA[d] = bsym[((size_t)b * NSYM_ + s) * DD_ + d];
        __syncthreads();
        float v = bv[d];
        for (int k = 0; k < DD_; ++k) v += bufA[k] * Wv[k * DD_ + d];
        bufB[d] = v; __syncthreads();
        float o = bo[d];
        for (int k = 0; k < DD_; ++k) o += bufB[k] * Wo[k * DD_ + d];
        __syncthreads();
        bufA[d] = o; __syncthreads();
        float l = bl[s * DD_ + d];
        for (int k = 0; k < DD_; ++k) l += bufA[k] * Wl[((size_t)s * DD_ + k) * DD_ + d];
        float sy = bufA[d] + l;
        ss[s][d] = sy; __syncthreads();
        float t = symb[d];
        for (int k = 0; k < DD_; ++k) t += ss[s][k] * symW[k * DD_ + d];
        red[d] = tanhf(t) * symu[d]; __syncthreads();
        for (int o2 = 64; o2 > 0; o2 >>= 1) {
            if (d < o2) red[d] += red[d + o2];
            __syncthreads();
        }
        if (d == 0) e2[s] = red[0];
        __syncthreads();
    }
    float m = fmaxf(fmaxf(e2[0], e2[1]), fmaxf(e2[2], e2[3]));
    float a0 = __expf(e2[0] - m), a1 = __expf(e2[1] - m);
    float a2 = __expf(e2[2] - m), a3 = __expf(e2[3] - m);
    float z = a0 + a1 + a2 + a3;
    syn_out[(size_t)b * DD_ + d] =
        (a0 * ss[0][d] + a1 * ss[1][d] + a2 * ss[2][d] + a3 * ss[3][d]) / z;
}

// ---------------- final: out[64][4000] = syn_out[64][128] @ z_tcm[4000][128]^T ----------------
__global__ void out_gemm_kernel(const float* __restrict__ syn_out,
                                const float* __restrict__ z_tcm, float* __restrict__ out) {
    int m0 = blockIdx.x * 16;                  // 4 row tiles
    int n0 = blockIdx.y * 128 + (threadIdx.x >> 5) * 16;
    int lane = threadIdx.x & 31;
    int half = lane >> 4, lr = lane & 15;
    int n = n0 + lr;
    bool nok = n < NT_;
    const float* ap = syn_out + (size_t)(m0 + lr) * DD_;
    const float* bp = z_tcm + (size_t)(nok ? n : 0) * DD_;   // B[k][n] = z_tcm[n][k]
    v8f c = {};
#pragma unroll 8
    for (int k0 = 0; k0 < DD_; k0 += 4) {
        int ka = k0 + 2 * half;
        v2f a, b;
        a.x = ap[ka]; a.y = ap[ka + 1];
        b.x = bp[ka];
        b.y = bp[ka + 1];
        c = __builtin_amdgcn_wmma_f32_16x16x4_f32(false, a, false, b, (short)0, c, false, false);
    }
    if (nok) {
#pragma unroll
        for (int g = 0; g < 8; ++g)
            out[(size_t)(m0 + g + 8 * half) * NT_ + n] = c[g];
    }
}

// ---------------- host ----------------
extern "C" void kernel_launch(void* const* d_in, const int* in_sizes, int n_in,
                              void* d_out, int out_size, void* d_ws, size_t ws_size,
                              hipStream_t stream) {
    const float* x_dis = (const float*)d_in[0];
    const float* x_syn = (const float*)d_in[1];
    const float* x_tcm = (const float*)d_in[2];
    const float* ds_Wsf = (const float*)d_in[3];
    const float* ds_bsf = (const float*)d_in[4];
    const float* ds_Wdf = (const float*)d_in[5];
    const float* ds_bdf = (const float*)d_in[6];
    const float* ds_se  = (const float*)d_in[7];
    const float* ds_de  = (const float*)d_in[8];
    const float* st_Wsf = (const float*)d_in[9];
    const float* st_bsf = (const float*)d_in[10];
    const float* st_Wdf = (const float*)d_in[11];
    const float* st_bdf = (const float*)d_in[12];
    const float* st_se  = (const float*)d_in[13];
    const float* st_de  = (const float*)d_in[14];
    const float* Wg  = (const float*)d_in[15];
    const float* bg  = (const float*)d_in[16];
    const float* att = (const float*)d_in[17];
    // d_in[18..20] syn_W/syn_b/syn_u feed only the dead b_syn branch -> unused
    const float* symW = (const float*)d_in[21];
    const float* symb = (const float*)d_in[22];
    const float* symu = (const float*)d_in[23];
    const float* Wv = (const float*)d_in[24];
    const float* bv = (const float*)d_in[25];
    const float* Wo = (const float*)d_in[26];
    const float* bo = (const float*)d_in[27];
    const float* Wl = (const float*)d_in[28];
    const float* bl = (const float*)d_in[29];
    const int* e1s = (const int*)d_in[30];
    const int* e1d = (const int*)d_in[31];
    const int* e2s = (const int*)d_in[32];
    const int* e2d = (const int*)d_in[33];
    const int* dmask = (const int*)d_in[34];
    // d_in[35] syn_mask: dead

    char* ws = (char*)d_ws;
    const size_t SZ_DS = (size_t)(ND_ + NS_) * DD_ * 4;      // 14,336,000
    const size_t SZ_ST = (size_t)(NS_ + NT_) * DD_ * 4;      //  6,144,000
    float* dsAcc = (float*)(ws + 0);
    float* dsEk  = (float*)(ws + SZ_DS);
    float* dsNew = (float*)(ws + 2 * SZ_DS);
    float* stAcc = (float*)(ws + 3 * SZ_DS);
    float* stEk  = (float*)(ws + 3 * SZ_DS + SZ_ST);
    float* stNew = (float*)(ws + 3 * SZ_DS + 2 * SZ_ST);
    size_t off = 3 * SZ_DS + 3 * SZ_ST;
    float* des    = (float*)(ws + off); off += (size_t)NSYM_ * ND_ * DD_ * 4;
    float* scores = (float*)(ws + off); off += (size_t)NSYM_ * ND_ * 4;
    float* smax   = (float*)(ws + off); off += (size_t)NSYM_ * BB_ * 4;
    float* sinv   = (float*)(ws + off); off += (size_t)NSYM_ * BB_ * 4;
    float* bsym   = (float*)(ws + off); off += (size_t)BB_ * NSYM_ * DD_ * 4;
    float* synout = (float*)(ws + off); off += (size_t)BB_ * DD_ * 4;
    float* degDs  = (float*)(ws + off); off += (size_t)(ND_ + NS_) * 4;
    float* degSt  = (float*)(ws + off); off += (size_t)(NS_ + NT_) * 4;
    float* Wmat   = dsEk;   // alias freed ds ping-pong buffers (20.5 MB < 28.7 MB)

    const long nDsElem = (long)(ND_ + NS_) * DD_;
    const long nStElem = (long)(NS_ + NT_) * DD_;

    // degrees -> dinv
    zero_kernel<<<cdiv(ND_ + NS_, 256), 256, 0, stream>>>(degDs, ND_ + NS_);
    zero_kernel<<<cdiv(NS_ + NT_, 256), 256, 0, stream>>>(degSt, NS_ + NT_);
    deg_kernel<<<cdiv(E1_, 256), 256, 0, stream>>>(e1s, e1d, E1_, ND_, degDs);
    deg_kernel<<<cdiv(E2_, 256), 256, 0, stream>>>(e2s, e2d, E2_, NS_, degSt);
    dinv_kernel<<<cdiv(ND_ + NS_, 256), 256, 0, stream>>>(degDs, ND_ + NS_);
    dinv_kernel<<<cdiv(NS_ + NT_, 256), 256, 0, stream>>>(degSt, NS_ + NT_);

    // e0 = [src_fc(x)+emb ; dst_fc(x)+emb]
    fc_gemm_kernel<384><<<ND_ / 16, 256, 0, stream>>>(x_dis, ds_Wsf, ds_bsf, ds_se, dsAcc);
    fc_gemm_kernel<256><<<NS_ / 16, 256, 0, stream>>>(x_syn, ds_Wdf, ds_bdf, ds_de, dsAcc + (size_t)ND_ * DD_);
    fc_gemm_kernel<256><<<NS_ / 16, 256, 0, stream>>>(x_syn, st_Wsf, st_bsf, st_se, stAcc);
    fc_gemm_kernel<103><<<NT_ / 16, 256, 0, stream>>>(x_tcm, st_Wdf, st_bdf, st_de, stAcc + (size_t)NS_ * DD_);

    hipMemcpyAsync(dsEk, dsAcc, SZ_DS, hipMemcpyDeviceToDevice, stream);
    hipMemcpyAsync(stEk, stAcc, SZ_ST, hipMemcpyDeviceToDevice, stream);

    // K=2 LightGCN rounds, ping-pong ek buffers, acc += ek
    int gDsE = cdiv(2L * E1_ * 32, 256), gStE = cdiv(2L * E2_ * 32, 256);
    int gDs4 = cdiv(nDsElem / 4, 256), gSt4 = cdiv(nStElem / 4, 256);
    zero4_kernel<<<gDs4, 256, 0, stream>>>((float4*)dsNew, nDsElem / 4);
    prop_kernel<<<gDsE, 256, 0, stream>>>(e1s, e1d, E1_, ND_, degDs, dsEk, dsNew);
    add4_kernel<<<gDs4, 256, 0, stream>>>((float4*)dsAcc, (const float4*)dsNew, nDsElem / 4);
    zero4_kernel<<<gDs4, 256, 0, stream>>>((float4*)dsEk, nDsElem / 4);
    prop_kernel<<<gDsE, 256, 0, stream>>>(e1s, e1d, E1_, ND_, degDs, dsNew, dsEk);
    add4_kernel<<<gDs4, 256, 0, stream>>>((float4*)dsAcc, (const float4*)dsEk, nDsElem / 4);
    scale4_kernel<<<gDs4, 256, 0, stream>>>((float4*)dsAcc, 1.f / 3.f, nDsElem / 4);

    zero4_kernel<<<gSt4, 256, 0, stream>>>((float4*)stNew, nStElem / 4);
    prop_kernel<<<gStE, 256, 0, stream>>>(e2s, e2d, E2_, NS_, degSt, stEk, stNew);
    add4_kernel<<<gSt4, 256, 0, stream>>>((float4*)stAcc, (const float4*)stNew, nStElem / 4);
    zero4_kernel<<<gSt4, 256, 0, stream>>>((float4*)stEk, nStElem / 4);
    prop_kernel<<<gStE, 256, 0, stream>>>(e2s, e2d, E2_, NS_, degSt, stNew, stEk);
    add4_kernel<<<gSt4, 256, 0, stream>>>((float4*)stAcc, (const float4*)stEk, nStElem / 4);
    scale4_kernel<<<gSt4, 256, 0, stream>>>((float4*)stAcc, 1.f / 3.f, nStElem / 4);

    const float* z_dis = dsAcc;                         // [ND,128]
    const float* z_tcm = stAcc + (size_t)NS_ * DD_;     // [NT,128]

    // gate + des + scores (WMMA fused with sigmoid epilogue)
    gate_kernel<<<dim3(ND_ / 16, NSYM_), 256, 0, stream>>>(z_dis, Wg, bg, att, des, scores);

    // masked softmax weights
    stats_kernel<<<dim3(BB_, NSYM_), 256, 0, stream>>>(scores, dmask, smax, sinv);
    wmat_kernel<<<cdiv((long)NSYM_ * BB_ * ND_, 256), 256, 0, stream>>>(scores, dmask, smax, sinv, Wmat);

    // b_sym = W @ des  (WMMA, K=20000)
    bsym_gemm_kernel<<<dim3(BB_ / 16, NSYM_), 256, 0, stream>>>(Wmat, des, bsym);

    // tail: sAtt + lins + sym_agg softmax
    tail_kernel<<<BB_, 128, 0, stream>>>(bsym, Wv, bv, Wo, bo, Wl, bl, symW, symb, symu, synout);

    // out = syn_out @ z_tcm^T
    out_gemm_kernel<<<dim3(BB_ / 16, cdiv(NT_, 128)), 256, 0, stream>>>(synout, z_tcm, (float*)d_out);
}